// ScaledNeuron_86732569575856
// MI455X (gfx1250) — compile-verified
//
#include <hip/hip_runtime.h>
#include <stdint.h>
#include <stddef.h>

// Spiking IF-neuron scan over T=8 (innermost, contiguous) for 4.19M neurons.
// Memory-bound streaming kernel: 268 MB total traffic -> ~11.5 us at 23.3 TB/s.
// Input staged through LDS with CDNA5 async global->LDS copies (ASYNCcnt),
// output streamed with non-temporal b128 stores.

typedef float v4f __attribute__((ext_vector_type(4)));

#define THREADS 256
#define T_STEPS 8

__global__ void __launch_bounds__(THREADS)
ScaledNeuron_if_scan(const float* __restrict__ xs,
                     float* __restrict__ out,
                     int n_neurons) {
  // 32 B of staging per lane: 256 * 32 B = 8 KB LDS per block.
  __shared__ v4f tile[THREADS * 2];

  const int tid = threadIdx.x;
  const int neuron = blockIdx.x * THREADS + tid;
  if (neuron >= n_neurons) return;

  // Global byte address of this neuron's 8 contiguous timesteps (32 B).
  // Consecutive lanes -> consecutive 32B chunks: fully coalesced b128 traffic.
  const uint64_t gaddr = (uint64_t)(uintptr_t)(xs + (size_t)neuron * 8u);

  // True LDS (addrspace 3) pointer to this lane's slot; its 32-bit value is
  // the LDS byte offset that the async instruction's VDST operand expects.
  typedef __attribute__((address_space(3))) v4f lds_v4f;
  lds_v4f* slot = (lds_v4f*)&tile[tid * 2];

  // CDNA5 async global->LDS copy path (tracked by ASYNCcnt, not LOADcnt).
  // INST_OFFSET is added to BOTH the LDS and global addresses (ISA 08, 4.4),
  // so offset:16 fetches the second float4 of the same neuron into the
  // adjacent LDS slot with the same base registers.
  asm volatile(
      "global_load_async_to_lds_b128 %0, %1, off\n\t"
      "global_load_async_to_lds_b128 %0, %1, off offset:16\n\t"
      "s_wait_asynccnt 0"
      :
      : "v"(slot), "v"(gaddr)
      : "memory");

  // Read back our own 32 B (ds_load_b128 x2; DScnt handled by compiler).
  v4f a = tile[tid * 2 + 0];
  v4f b = tile[tid * 2 + 1];

  // Integrate-and-fire scan, fully in VGPRs (branchless).
  float x[T_STEPS] = {a.x, a.y, a.z, a.w, b.x, b.y, b.z, b.w};
  float s[T_STEPS];
  float v = 0.5f;  // t==0 pre-charge per reference
#pragma unroll
  for (int t = 0; t < T_STEPS; ++t) {
    v += x[t];                               // neuronal_charge
    float sp = (v >= 1.0f) ? 1.0f : 0.0f;    // fire (v - V_TH >= 0)
    v -= sp;                                 // soft reset
    s[t] = sp;                               // SCALE == 1
  }

  // Streamed once -> non-temporal b128 stores (th:NT), avoid L2 pollution.
  v4f o0 = {s[0], s[1], s[2], s[3]};
  v4f o1 = {s[4], s[5], s[6], s[7]};
  v4f* op = (v4f*)(out + (size_t)neuron * 8u);
  __builtin_nontemporal_store(o0, op + 0);
  __builtin_nontemporal_store(o1, op + 1);
}

extern "C" void kernel_launch(void* const* d_in, const int* in_sizes, int n_in,
                              void* d_out, int out_size, void* d_ws, size_t ws_size,
                              hipStream_t stream) {
  (void)in_sizes; (void)n_in; (void)d_ws; (void)ws_size;
  const float* xs = (const float*)d_in[0];   // [B,C,H,W,T] f32, T innermost
  float* out = (float*)d_out;                // same shape/dtype
  const int n_neurons = out_size / T_STEPS;  // 4,194,304
  const int blocks = (n_neurons + THREADS - 1) / THREADS;  // 16384
  ScaledNeuron_if_scan<<<blocks, THREADS, 0, stream>>>(xs, out, n_neurons);
}